// RuleBasedLM_40776419508496
// MI455X (gfx1250) — compile-verified
//
#include <hip/hip_runtime.h>

// ---------------------------------------------------------------------------
// RuleBasedLM for MI455X (gfx1250, wave32).
// Outputs (concatenated in d_out): logits [L,16] f32, attn [L,L] f32, h [L,64] f32.
//   * attn (256 MB) dominates: NT b128 streaming stores, ~12 us roofline @23.3TB/s
//   * h done as one-hot x emb GEMM via V_WMMA_F32_16X16X4_F32 (bit-exact gather)
//   * emb table (4 KB) staged to LDS once per workgroup via TDM
//     (tensor_load_to_lds + s_wait_tensorcnt); B operands come from ds_load.
//     An asm escape/clobber keeps the compiler from folding the LDS reads
//     (it cannot see the TDM write, so the array would otherwise look store-free).
// ---------------------------------------------------------------------------

#define NEGV (-1e9f)

typedef __attribute__((ext_vector_type(2))) float        v2f;
typedef __attribute__((ext_vector_type(4))) float        v4f;
typedef __attribute__((ext_vector_type(8))) float        v8f;
typedef __attribute__((ext_vector_type(4))) unsigned int v4u;
typedef __attribute__((ext_vector_type(4))) int          v4i;
typedef __attribute__((ext_vector_type(8))) int          v8i;

// ---------------- logits: one thread per sequence position -----------------
__global__ void rb_logits_kernel(const int* __restrict__ ids,
                                 float* __restrict__ logits, int L) {
    int l = blockIdx.x * blockDim.x + threadIdx.x;
    if (l >= L) return;
    float row[16];
    if (l == 0) {
#pragma unroll
        for (int t = 0; t < 16; ++t) row[t] = NEGV;
    } else {
        int prev = ids[l - 1];
        bool act = (prev == 7) | (prev == 8) | (prev == 9);
#pragma unroll
        for (int t = 0; t < 16; ++t) {
            bool pref    = (t == 10) | (t == 11);
            bool special = (t <= 2);
            row[t] = act ? (pref ? 5.0f : NEGV)
                         : (pref ? -5.0f : (special ? NEGV : 1.0f));
        }
    }
    v4f* out = (v4f*)(logits + (size_t)l * 16);
#pragma unroll
    for (int q = 0; q < 4; ++q) {
        v4f w = { row[4 * q + 0], row[4 * q + 1], row[4 * q + 2], row[4 * q + 3] };
        out[q] = w;
    }
}

// ---------------- attn: one workgroup per row, NT b128 streaming ------------
__global__ void rb_attn_kernel(const int* __restrict__ ids,
                               float* __restrict__ attn, int L) {
    int i = blockIdx.x;
    int id_i = ids[i];
    bool pref = (id_i == 10) | (id_i == 11);
    bool cond = false;
    if (pref && i > 0) {
        int p = ids[i - 1];
        cond = (p == 7) | (p == 8) | (p == 9);
    }
    float invL = 1.0f / (float)L;
    float each = 0.2f / (float)(L - 1);
    float base = cond ? each : invL;

    v4f* row = (v4f*)(attn + (size_t)i * L);
    int nvec = L >> 2;
    if (!cond) {
        v4f w = { base, base, base, base };
        for (int q = threadIdx.x; q < nvec; q += blockDim.x)
            __builtin_nontemporal_store(w, &row[q]);
    } else {
        int tgt = i - 1;  // column that receives 0.8
        for (int q = threadIdx.x; q < nvec; q += blockDim.x) {
            v4f w = { base, base, base, base };
            int c0 = q << 2;
            if (tgt >= c0 && tgt < c0 + 4) w[tgt - c0] = 0.8f;
            __builtin_nontemporal_store(w, &row[q]);
        }
    }
}

// ---- TDM: DMA the 16x64 f32 emb table (4 KB) into LDS offset 0 -------------
// D# group0: count=1 (valid user desc), lds_addr=0, 57-bit global addr, type=2.
// D# group1: data_size=4B (code 2), 1D tile: tensor_dim0=tile_dim0=1024 elems,
//            tensor_dim0_stride=1024, tensor_dim1=1, everything else 0.
// Groups 2/3 zero (<=2D tensor). Tracked by TENSORcnt.
#if __has_builtin(__builtin_amdgcn_tensor_load_to_lds)
#define RB_HAVE_TDM 1
__device__ __forceinline__ void tdm_load_emb_to_lds(const float* emb) {
    unsigned long long ga = (unsigned long long)emb;
    v4u g0 = { 1u,                                   // count=1, user descriptor
               0u,                                   // lds_addr = 0 (only LDS array)
               (unsigned)(ga & 0xFFFFFFFFu),
               (unsigned)(((ga >> 32) & 0x01FFFFFFu) | 0x80000000u) }; // type=2
    v8i g1 = { (int)0x00020000,                      // data_size=2 (4 bytes)
               (int)0x04000000,                      // tensor_dim0[15:0]=1024 << 16
               (int)0x00010000,                      // tensor_dim1[15:0]=1   << 16
               (int)0x04000000,                      // tile_dim0=1024        << 16
               0,                                    // tile_dim1=0, tile_dim2=0
               0x400,                                // tensor_dim0_stride=1024
               0, 0 };
    v4i z4 = { 0, 0, 0, 0 };
#if defined(__clang_major__) && (__clang_major__ >= 23)
    v8i z8 = { 0, 0, 0, 0, 0, 0, 0, 0 };
    __builtin_amdgcn_tensor_load_to_lds(g0, g1, z4, z4, z8, 0);
#else
    __builtin_amdgcn_tensor_load_to_lds(g0, g1, z4, z4, 0);
#endif
    __builtin_amdgcn_s_wait_tensorcnt(0);            // TENSORcnt == 0
}
#else
#define RB_HAVE_TDM 0
#endif

// ---------------- h: one-hot gather as WMMA f32 16x16x4 GEMM ----------------
// One wave per 16-token tile. h_tile[16x64] = onehot[16x16(vocab)] x emb[16x64].
// K=16 vocab is covered by 4 WMMA K-steps of 4; 4 chunks of 16 emb dims.
__global__ void rb_emb_wmma_kernel(const int* __restrict__ ids,
                                   const float* __restrict__ emb,
                                   float* __restrict__ h, int L) {
    __shared__ float lds_emb[16 * 64];               // must sit at LDS offset 0

    // Stage emb into LDS: wave 0 issues the TDM DMA, everyone barriers.
#if RB_HAVE_TDM
    if (threadIdx.x < 32) tdm_load_emb_to_lds(emb);
#else
    if (threadIdx.x < 32)
        for (int t = threadIdx.x; t < 16 * 64; t += 32) lds_emb[t] = emb[t];
#endif
    __syncthreads();

    // Escape lds_emb's address into a side-effecting asm with a memory clobber.
    // The TDM write is invisible to the compiler; without this, "no stores to
    // lds_emb" lets GlobalOpt fold every ds_load to undef (seen in round 2:
    // ds=0 and the four chunk GEMMs CSE'd into one).
    {
        float* escape = lds_emb;
        asm volatile("" : "+v"(escape) :: "memory");
    }

    int lane = threadIdx.x & 31;
    int wave = threadIdx.x >> 5;
    int tile = blockIdx.x * (blockDim.x >> 5) + wave;
    int ntiles = L >> 4;
    if (tile >= ntiles) return;          // wave-uniform; EXEC stays all-ones

    int mrow = lane & 15;                // M row (A/C) and N col (B/D)
    int half = lane >> 4;                // 0: K offset {0,1}; 1: K offset {2,3}
    int my_id = ids[tile * 16 + mrow];

#pragma unroll
    for (int c = 0; c < 4; ++c) {        // 16-wide emb-dim chunk
        v8f acc = {};
#pragma unroll
        for (int kk = 0; kk < 4; ++kk) { // vocab chunk [kk*4, kk*4+4)
            int k0 = kk * 4 + half * 2;
            v2f a, b;
            // A 16x4 f32 layout: lanes 0-15 M=0..15 with VGPR0=K0,VGPR1=K1;
            // lanes 16-31 same M with K2,K3.
            a.x = (my_id == k0 + 0) ? 1.0f : 0.0f;
            a.y = (my_id == k0 + 1) ? 1.0f : 0.0f;
            // B 4x16 f32: lane indexes N, VGPR/half-wave index K (mirrors A).
            b.x = lds_emb[(k0 + 0) * 64 + c * 16 + mrow];
            b.y = lds_emb[(k0 + 1) * 64 + c * 16 + mrow];
            acc = __builtin_amdgcn_wmma_f32_16x16x4_f32(
                false, a, false, b, (short)0, acc, false, false);
        }
        // D 16x16 f32 layout: VGPR v -> M = v + half*8, N = lane&15
#pragma unroll
        for (int v = 0; v < 8; ++v) {
            int M = v + half * 8;
            h[(size_t)(tile * 16 + M) * 64 + c * 16 + mrow] = acc[v];
        }
    }
}

extern "C" void kernel_launch(void* const* d_in, const int* in_sizes, int n_in,
                              void* d_out, int out_size, void* d_ws, size_t ws_size,
                              hipStream_t stream) {
    const int*   ids = (const int*)d_in[0];     // [1, L] int32
    const float* emb = (const float*)d_in[1];   // [16, 64] f32
    int L = in_sizes[0];                        // 8192

    float* logits = (float*)d_out;              // [L,16]
    float* attn   = logits + (size_t)L * 16;    // [L,L]
    float* h      = attn + (size_t)L * L;       // [L,64]

    rb_logits_kernel<<<(L + 255) / 256, 256, 0, stream>>>(ids, logits, L);
    rb_attn_kernel<<<L, 256, 0, stream>>>(ids, attn, L);

    int ntiles = L >> 4;                        // 512 waves of WMMA work
    int wavesPerBlock = 8;                      // 256 threads = 8 waves
    int blocks = (ntiles + wavesPerBlock - 1) / wavesPerBlock;
    rb_emb_wmma_kernel<<<blocks, wavesPerBlock * 32, 0, stream>>>(ids, emb, h, L);
}